// NewtonSORJit_22479858827503
// MI455X (gfx1250) — compile-verified
//
#include <hip/hip_runtime.h>

// Newton-SOR for 3-species stiff kinetics, B=524288 rows, fp32.
// Roofline: ~23 MB traffic (~1us @ 23.3 TB/s) vs ~1 GFLOP dependent scalar
// fp32 -> VALU-latency bound; no GEMM structure, so WMMA is inapplicable.
// CDNA5 lever: per-wave async global->LDS staging (ASYNCcnt path) of the
// 28B-strided AoS rows; saddr-form async loads share one vaddr/ldsaddr VGPR
// and the instruction offset advances BOTH LDS and global address
// (CDNA5 ISA) -> 7 straight-line VMEM issues per wave.
// Readback is a SINGLE path from LDS (tail waves also stage through LDS,
// exec-masked) so the compiler can lower it to ds_load, not flat_load.

#define NS_TOL2   (1e-12f)   // TOL^2, compare ||F||^2 (avoids sqrt)
#define NS_MAXIT  50
#define NS_CMIN  (-1.0f)
#define NS_CMAX  ( 2.0f)

__device__ __forceinline__ float fast_rcp(float x) {
  // v_rcp_f32 (~1 ulp) + one Newton-Raphson refinement
  float r = __builtin_amdgcn_rcpf(x);
  return r * (2.0f - x * r);
}

__device__ __forceinline__ void eval_F(float y0, float y1, float y2,
                                       float yo0, float yo1, float yo2,
                                       float h, float k1, float k2, float k3,
                                       float& F0, float& F1, float& F2) {
  float y12  = y1 * y2;
  float y1sq = y1 * y1;
  F0 = yo0 + h * (-k1 * y0 + k3 * y12) - y0;
  F1 = yo1 + h * ( k1 * y0 - k2 * y1sq - k3 * y12) - y1;
  F2 = yo2 + h * ( k2 * y1sq) - y2;
}

__global__ __launch_bounds__(256) void newton_sor_kernel(
    const float* __restrict__ x,      // (n, 7) row-major
    const float* __restrict__ omega,  // (n, 1)
    float* __restrict__ out,          // (n, 3) row-major
    int n) {
  // 8 wave32 waves/block * 224 floats = 7168 B of LDS; per-wave tiles, so no
  // workgroup barrier needed (each wave reads only what it issued).
  __shared__ float sh[8 * 224];

  const int lane = (int)(threadIdx.x & 31u);
  const int wv   = (int)(threadIdx.x >> 5u);
  const int gid  = (int)(blockIdx.x * blockDim.x + threadIdx.x);
  // Wave-uniform first row of this wave (readfirstlane -> SGPR, scalar branch)
  const int wb = __builtin_amdgcn_readfirstlane(
      (int)(blockIdx.x * blockDim.x) + (wv << 5));

  // ---- async global -> LDS staging (ASYNCcnt path) ----
  // ISA 10.2: low 32 bits of a generic LDS address == LDS byte offset.
  const unsigned ldsOff = (unsigned)(size_t)(&sh[wv * 224]) +
                          ((unsigned)lane << 2);
  const unsigned vOff   = (unsigned)lane << 2;              // per-lane byte off
  const float* base     = x + (size_t)(unsigned)wb * 7u;    // uniform -> saddr

#define NS_ASYNC_LD(OFF)                                                \
  asm volatile("global_load_async_to_lds_b32 %0, %1, %2 offset:" #OFF   \
               :: "v"(ldsOff), "v"(vOff), "s"(base) : "memory")

  if (wb + 32 <= n) {
    // fast path: full wave, 7 straight-line coalesced issues (896 B = 32 rows)
    NS_ASYNC_LD(0);   NS_ASYNC_LD(128); NS_ASYNC_LD(256); NS_ASYNC_LD(384);
    NS_ASYNC_LD(512); NS_ASYNC_LD(640); NS_ASYNC_LD(768);
  } else {
    // rare tail wave: per-lane exec-masked issues, same LDS layout
    const int limF = (n - wb) * 7;              // valid floats in this tile
    if (lane +   0 < limF) NS_ASYNC_LD(0);
    if (lane +  32 < limF) NS_ASYNC_LD(128);
    if (lane +  64 < limF) NS_ASYNC_LD(256);
    if (lane +  96 < limF) NS_ASYNC_LD(384);
    if (lane + 128 < limF) NS_ASYNC_LD(512);
    if (lane + 160 < limF) NS_ASYNC_LD(640);
    if (lane + 192 < limF) NS_ASYNC_LD(768);
  }
#undef NS_ASYNC_LD

#if __has_builtin(__builtin_amdgcn_s_wait_asynccnt)
  __builtin_amdgcn_s_wait_asynccnt(0);
  asm volatile("" ::: "memory");         // compiler barrier before LDS readback
#else
  asm volatile("s_wait_asynccnt 0" ::: "memory");
#endif

  // Single readback path, straight from the __shared__ array -> ds_load.
  // Stride 7 floats: gcd(7,64)=1 -> bank-conflict free across 32 lanes.
  const int rbase = wv * 224 + lane * 7;
  const float yo0 = sh[rbase + 0];
  const float yo1 = sh[rbase + 1];
  const float yo2 = sh[rbase + 2];
  const float h   = sh[rbase + 3];
  const float k1  = sh[rbase + 4];
  const float k2  = sh[rbase + 5];
  const float k3  = sh[rbase + 6];

  if (gid >= n) return;                  // tail lanes bail after readback

  const float om = omega[gid];

  float y0 = yo0, y1 = yo1, y2 = yo2;

  const float hk1  = h * k1;                 // == J10, loop-invariant
  const float rJ00 = fast_rcp(-hk1 - 1.0f);  // loop-invariant reciprocal

  float F0, F1, F2;
  eval_F(y0, y1, y2, yo0, yo1, yo2, h, k1, k2, k3, F0, F1, F2);

#pragma unroll 1
  for (int it = 0; it < NS_MAXIT; ++it) {
    float ss  = F0 * F0 + F1 * F1 + F2 * F2;
    bool  act = ss > NS_TOL2;
#if __has_builtin(__builtin_amdgcn_ballot_w32)
    if (__builtin_amdgcn_ballot_w32(act) == 0u) break;   // wave converged
#else
    if (!__any((int)act)) break;
#endif
    if (act) {
      float J11 = h * (-2.0f * k2 * y1 - k3 * y2) - 1.0f;
      float J21 = 2.0f * h * k2 * y1;
      float dx0 = F0 * rJ00;
      float dx1 = (F1 - om * hk1 * dx0) * fast_rcp(J11);
      float dx2 = om * J21 * dx1 - F2;                    // divide by -1.0
      y0 = fminf(fmaxf(y0 - om * dx0, NS_CMIN), NS_CMAX);
      y1 = fminf(fmaxf(y1 - om * dx1, NS_CMIN), NS_CMAX);
      y2 = fminf(fmaxf(y2 - om * dx2, NS_CMIN), NS_CMAX);
      eval_F(y0, y1, y2, yo0, yo1, yo2, h, k1, k2, k3, F0, F1, F2);
    }
  }

  out[3 * gid + 0] = y0;
  out[3 * gid + 1] = y1;
  out[3 * gid + 2] = y2;
}

extern "C" void kernel_launch(void* const* d_in, const int* in_sizes, int n_in,
                              void* d_out, int out_size, void* d_ws, size_t ws_size,
                              hipStream_t stream) {
  (void)n_in; (void)out_size; (void)d_ws; (void)ws_size;
  const float* x     = (const float*)d_in[0];   // (B,7) fp32
  const float* omega = (const float*)d_in[1];   // (B,1) fp32
  float* out         = (float*)d_out;           // (B,3) fp32

  const int n = in_sizes[0] / 7;
  const int block = 256;                         // 8 wave32 waves / block
  const int grid  = (n + block - 1) / block;     // 2048 blocks for B=524288
  newton_sor_kernel<<<grid, block, 0, stream>>>(x, omega, out, n);
}